// TransferModel_87514253623859
// MI455X (gfx1250) — compile-verified
//
#include <hip/hip_runtime.h>

// Problem constants (fixed by the reference).
#define LSEQ 512
#define NMUT 65536
#define DHID 128
#define VOC  21
#define KC   9

typedef __attribute__((ext_vector_type(2))) float v2f;
typedef __attribute__((ext_vector_type(8))) float v8f;
typedef __attribute__((ext_vector_type(4))) int   v4i;

#if defined(__HIP_DEVICE_COMPILE__) && __has_builtin(__builtin_amdgcn_wmma_f32_16x16x4_f32)
#define USE_WMMA 1
#else
#define USE_WMMA 0
#endif

__device__ __forceinline__ int imin(int a, int b) { return a < b ? a : b; }

// One 16-row x (nWaves*16)-col layer: out = opt_relu(in) @ B^T + bias,
// B row-major [realN][Kdim] with element stride sb (sb=9 grabs Conv1d center tap).
// Columns >= realN are zero-padded. One wave per 16-col tile, f32 WMMA 16x16x4.
__device__ __forceinline__ void layer_tile(
    const float* __restrict__ in, int Kdim, int ldin,
    const float* __restrict__ Bg, int ldb, int sb, int realN, int nWaves,
    const float* __restrict__ bias, bool relu,
    float* __restrict__ out, int ldout, int tid)
{
#if USE_WMMA
  int lane = tid & 31;
  int wv = tid >> 5;
  if (wv < nWaves) {                 // uniform per wave: EXEC stays all-1s for WMMA
    int lg = lane >> 4, li = lane & 15;
    int nn = (wv << 4) + li;
    bool nok = nn < realN;
    const float* bp = Bg + (size_t)(nok ? nn : 0) * ldb;
    v8f acc = {};
    for (int k0 = 0; k0 < Kdim; k0 += 4) {
      int ka = k0 + (lg << 1);
      // A frag (16x4): reg r = A[m=li][k=ka+r]
      float a0 = in[li * ldin + ka];
      float a1 = in[li * ldin + ka + 1];
      if (relu) { a0 = fmaxf(a0, 0.f); a1 = fmaxf(a1, 0.f); }
      // B frag (4x16): reg r = B[k=ka+r][n=li] = W[n][k]
      float b0 = nok ? bp[(size_t)ka * sb] : 0.f;
      float b1 = nok ? bp[(size_t)(ka + 1) * sb] : 0.f;
      v2f av = {a0, a1};
      v2f bv = {b0, b1};
      acc = __builtin_amdgcn_wmma_f32_16x16x4_f32(false, av, false, bv,
                                                  (short)0, acc, false, false);
    }
    float bn = nok ? bias[nn] : 0.f;
#pragma unroll
    for (int r = 0; r < 8; ++r) {    // D frag: reg r = C[m=r+8*lg][n=li]
      int m = r + (lg << 3);
      out[m * ldout + nn] = acc[r] + bn;
    }
  }
#else
  int Nout = nWaves << 4;
  for (int idx = tid; idx < 16 * Nout; idx += 256) {
    int m = idx / Nout, n = idx % Nout;
    float acc = 0.f;
    if (n < realN) {
      for (int k = 0; k < Kdim; ++k) {
        float a = in[m * ldin + k];
        if (relu) a = fmaxf(a, 0.f);
        acc += a * Bg[(size_t)n * ldb + (size_t)k * sb];
      }
      acc += bias[n];
    }
    out[m * ldout + n] = acc;
  }
#endif
}

// Precompute T[pos][0..31] = MLP(all_input[pos] @ Wf[:,:,4]^T + bf) for all 512 positions.
// One workgroup (8 waves) per 16-row strip; layers ping-pong through LDS (padded ld
// avoids 16-way bank conflicts on the row-major A reads).
__global__ __launch_bounds__(256) void mlp_table_kernel(
    const float* __restrict__ all_input,
    const float* __restrict__ Wf, const float* __restrict__ bf,
    const float* __restrict__ W1, const float* __restrict__ b1,
    const float* __restrict__ W2, const float* __restrict__ b2,
    const float* __restrict__ W3, const float* __restrict__ b3,
    const float* __restrict__ W4, const float* __restrict__ b4,
    float* __restrict__ T)
{
  __shared__ float bufA[16 * 132];
  __shared__ float bufB[16 * 132];
  int tid = threadIdx.x;
  int m0 = blockIdx.x << 4;

  for (int i = tid; i < 16 * 128; i += 256) {
    int r = i >> 7, c = i & 127;
    bufA[r * 132 + c] = all_input[(m0 + r) * 128 + c];
  }
  __syncthreads();
  // feat = x @ Wfc^T + bf          (no relu on input)
  layer_tile(bufA, 128, 132, Wf + (KC / 2), DHID * KC, KC, 128, 8, bf, false, bufB, 132, tid);
  __syncthreads();
  // h1 = relu(feat) @ W1^T + b1
  layer_tile(bufB, 128, 132, W1, 128, 1, 128, 8, b1, true, bufA, 132, tid);
  __syncthreads();
  // h2 = relu(h1) @ W2^T + b2      (128 -> 64)
  layer_tile(bufA, 128, 132, W2, 128, 1, 64, 4, b2, true, bufB, 68, tid);
  __syncthreads();
  // h3 = relu(h2) @ W3^T + b3      (64 -> 32)
  layer_tile(bufB, 64, 68, W3, 64, 1, 32, 2, b3, true, bufA, 36, tid);
  __syncthreads();
  // logits = relu(h3) @ W4^T + b4  (32 -> 21, padded to 32)
  layer_tile(bufA, 32, 36, W4, 32, 1, VOC, 2, b4, true, bufB, 36, tid);
  __syncthreads();
  for (int i = tid; i < 16 * 32; i += 256) {
    int r = i >> 5, c = i & 31;
    T[(m0 + r) * 32 + c] = bufB[r * 36 + c];
  }
}

// Bandwidth-bound pass: one wave per mutant row (512 int32 = 2KB, 4 x b128 per lane,
// nontemporal: single-use stream, don't churn L2). First mismatch via min-reduce of
// (idx<<8)|char keys across the wave32. Then 2-float lookup in the 64KB logits table.
__global__ __launch_bounds__(256) void scan_kernel(
    const int* __restrict__ muts, const int* __restrict__ wt,
    const float* __restrict__ T, const float* __restrict__ nrgs,
    const float* __restrict__ Wd,
    float* __restrict__ preds, float* __restrict__ reals)
{
  __shared__ int s_wt[LSEQ];
  int tid = threadIdx.x;
  for (int i = tid; i < LSEQ; i += 256) s_wt[i] = wt[i];
  __syncthreads();

  int lane = tid & 31;
  int wv = tid >> 5;
  int n = (blockIdx.x << 3) + wv;

  const v4i* row = (const v4i*)(muts + (size_t)n * LSEQ);
  int key = 0x7fffffff;
#pragma unroll
  for (int it = 0; it < 4; ++it) {
    v4i v = __builtin_nontemporal_load(row + it * 32 + lane);
    int base = (it << 7) + (lane << 2);
    if (v.x != s_wt[base + 0]) key = imin(key, ((base + 0) << 8) | v.x);
    if (v.y != s_wt[base + 1]) key = imin(key, ((base + 1) << 8) | v.y);
    if (v.z != s_wt[base + 2]) key = imin(key, ((base + 2) << 8) | v.z);
    if (v.w != s_wt[base + 3]) key = imin(key, ((base + 3) << 8) | v.w);
  }
#pragma unroll
  for (int o = 16; o > 0; o >>= 1) key = imin(key, __shfl_xor(key, o, 32));

  if (lane == 0) {
    float p = 0.f, r = 0.f;
    if (key != 0x7fffffff) {                 // valid (mismatch exists)
      int pos = key >> 8;
      int mc = key & 0xff;                   // mutant char at first mismatch
      int wc = s_wt[pos];                    // wildtype char
      p = Wd[0] * (T[(pos << 5) + wc] - T[(pos << 5) + mc]);  // bd cancels
      r = nrgs[n];
    }
    preds[n] = p;
    reals[n] = r;
  }
}

extern "C" void kernel_launch(void* const* d_in, const int* in_sizes, int n_in,
                              void* d_out, int out_size, void* d_ws, size_t ws_size,
                              hipStream_t stream) {
  (void)in_sizes; (void)n_in; (void)out_size; (void)ws_size;
  const float* all_input = (const float*)d_in[0];
  const int*   wt        = (const int*)d_in[1];
  const int*   muts      = (const int*)d_in[2];
  const float* nrgs      = (const float*)d_in[3];
  const float* Wf        = (const float*)d_in[4];
  const float* bf        = (const float*)d_in[5];
  // d_in[6] (Wa), d_in[7] (ba): dead — softmax over a length-1 axis is 1.
  const float* W1 = (const float*)d_in[8];
  const float* b1 = (const float*)d_in[9];
  const float* W2 = (const float*)d_in[10];
  const float* b2 = (const float*)d_in[11];
  const float* W3 = (const float*)d_in[12];
  const float* b3 = (const float*)d_in[13];
  const float* W4 = (const float*)d_in[14];
  const float* b4 = (const float*)d_in[15];
  const float* Wd = (const float*)d_in[16];
  // d_in[17] (bd): cancels in the wc-mc difference.

  float* T     = (float*)d_ws;            // [512][32] logits table, 64 KB
  float* preds = (float*)d_out;           // [65536]
  float* reals = preds + NMUT;            // [65536]

  mlp_table_kernel<<<LSEQ / 16, 256, 0, stream>>>(all_input, Wf, bf, W1, b1,
                                                  W2, b2, W3, b3, W4, b4, T);
  scan_kernel<<<NMUT / 8, 256, 0, stream>>>(muts, wt, T, nrgs, Wd, preds, reals);
}